// SQLGroupByMappingModel_24541443129981
// MI455X (gfx1250) — compile-verified
//
#include <hip/hip_runtime.h>
#include <hip/hip_bf16.h>

#define GROUPS 1024
#define NCOPIES 8
#define BLOCK 256
#define MAIN_GRID 1024
#define WAVES_PER_BLOCK (BLOCK / 32)
#define CHUNK 128  /* elements per wave-chunk: 32 lanes x 4 */

#define GAS __attribute__((address_space(1)))
#define LAS __attribute__((address_space(3)))

typedef int v4i __attribute__((vector_size(16)));

#if __has_builtin(__builtin_amdgcn_global_load_async_to_lds_b128)
#define HAVE_ASYNC 1
#else
#define HAVE_ASYNC 0
#endif

#if __has_builtin(__builtin_amdgcn_s_wait_asynccnt)
#define ASYNC_WAIT(n)                                      \
  do {                                                     \
    __builtin_amdgcn_s_wait_asynccnt((unsigned short)(n)); \
    asm volatile("" ::: "memory");                         \
  } while (0)
#else
#define ASYNC_WAIT(n) asm volatile("s_wait_asynccnt %0" ::"n"(n) : "memory")
#endif

// Monotonic uint encoding of float: preserves ordering for integer min/max.
__device__ __forceinline__ unsigned ordf(float f) {
  unsigned u = __float_as_uint(f);
  return (u & 0x80000000u) ? ~u : (u | 0x80000000u);
}
__device__ __forceinline__ float unordf(unsigned e) {
  return (e & 0x80000000u) ? __uint_as_float(e & 0x7FFFFFFFu)
                           : __uint_as_float(~e);
}

__global__ void gb_init(float* __restrict__ gsum, unsigned* __restrict__ gcnt,
                        unsigned* __restrict__ gmn, unsigned* __restrict__ gmx) {
  int i = blockIdx.x * blockDim.x + threadIdx.x;
  if (i < NCOPIES * GROUPS) {
    gsum[i] = 0.0f;
    gcnt[i] = 0u;
    gmn[i] = 0xFFFFFFFFu;  // identity for ordered-uint min
    gmx[i] = 0u;           // identity for ordered-uint max
  }
}

__global__ __launch_bounds__(BLOCK) void gb_main(
    const int* __restrict__ keys, const float* __restrict__ vals,
    float* __restrict__ gsum, unsigned* __restrict__ gcnt,
    unsigned* __restrict__ gmn, unsigned* __restrict__ gmx, int n) {
  __shared__ float lsum[GROUPS];
  __shared__ unsigned lcnt[GROUPS];
  __shared__ unsigned lmn[GROUPS];
  __shared__ unsigned lmx[GROUPS];
#if HAVE_ASYNC
  __shared__ int4 skey[2][WAVES_PER_BLOCK][32];
  __shared__ float4 sval[2][WAVES_PER_BLOCK][32];
#endif

  const int tid = threadIdx.x;
  for (int i = tid; i < GROUPS; i += BLOCK) {
    lsum[i] = 0.0f;
    lcnt[i] = 0u;
    lmn[i] = 0xFFFFFFFFu;
    lmx[i] = 0u;
  }
  __syncthreads();

#define PROC1(kk, vv)                                            \
  do {                                                           \
    unsigned g = (unsigned)((GROUPS - 1) - (kk)) & (GROUPS - 1); \
    atomicAdd(&lsum[g], (vv));                                   \
    atomicAdd(&lcnt[g], 1u);                                     \
    unsigned e = ordf(vv);                                       \
    atomicMin(&lmn[g], e);                                       \
    atomicMax(&lmx[g], e);                                       \
  } while (0)

  const unsigned numFull = (unsigned)n / CHUNK;
  {  // tail (n not multiple of CHUNK); no-op for N = 16M
    unsigned tail = (unsigned)n - numFull * CHUNK;
    unsigned t = blockIdx.x * BLOCK + tid;
    if (t < tail) {
      int k = keys[numFull * CHUNK + t];
      float v = vals[numFull * CHUNK + t];
      PROC1(k, v);
    }
  }

  const unsigned lane = tid & 31u;
  const unsigned widx = (unsigned)tid >> 5;
  const unsigned gwave = blockIdx.x * WAVES_PER_BLOCK + widx;
  const unsigned step = gridDim.x * WAVES_PER_BLOCK;

#if HAVE_ASYNC
  auto issue = [&](unsigned c, int b) {
    // lane copies 16B of keys and 16B of vals of chunk c into this wave's slot
    GAS char* gk = (GAS char*)keys + (size_t)c * (CHUNK * 4) + lane * 16;
    GAS char* gv = (GAS char*)vals + (size_t)c * (CHUNK * 4) + lane * 16;
    LAS char* lk = (LAS char*)&skey[b][widx][lane];
    LAS char* lv = (LAS char*)&sval[b][widx][lane];
    __builtin_amdgcn_global_load_async_to_lds_b128((GAS v4i*)gk, (LAS v4i*)lk, 0, 0);
    __builtin_amdgcn_global_load_async_to_lds_b128((GAS v4i*)gv, (LAS v4i*)lv, 0, 0);
  };

  unsigned c = gwave;
  if (c < numFull) {
    issue(c, 0);
    unsigned c1 = c + step;
    if (c1 < numFull) issue(c1, 1);
    int buf = 0;
    while (c < numFull) {
      if (c + step < numFull) {
        ASYNC_WAIT(2);  // current buffer's pair retired; next pair in flight
      } else {
        ASYNC_WAIT(0);
      }
      int4 kk = skey[buf][widx][lane];
      float4 vv = sval[buf][widx][lane];
      PROC1(kk.x, vv.x);
      PROC1(kk.y, vv.y);
      PROC1(kk.z, vv.z);
      PROC1(kk.w, vv.w);
      unsigned cn = c + 2u * step;
      if (cn < numFull) issue(cn, buf);  // safe: ds loads above already consumed
      buf ^= 1;
      c += step;
    }
  }
#else
  for (unsigned c = gwave; c < numFull; c += step) {
    int4 kk = *(const int4*)(keys + (size_t)c * CHUNK + lane * 4);
    float4 vv = *(const float4*)(vals + (size_t)c * CHUNK + lane * 4);
    PROC1(kk.x, vv.x);
    PROC1(kk.y, vv.y);
    PROC1(kk.z, vv.z);
    PROC1(kk.w, vv.w);
  }
#endif

  __syncthreads();

  // Flush WG-private histogram into one of NCOPIES global accumulator copies.
  const unsigned copy = blockIdx.x & (NCOPIES - 1);
  for (int i = tid; i < GROUPS; i += BLOCK) {
    unsigned cc = lcnt[i];
    if (cc) {
      unsigned o = copy * GROUPS + (unsigned)i;
      atomicAdd(&gsum[o], lsum[i]);
      atomicAdd(&gcnt[o], cc);
      atomicMin(&gmn[o], lmn[i]);
      atomicMax(&gmx[o], lmx[i]);
    }
  }
#undef PROC1
}

__global__ void gb_final(const float* __restrict__ gsum,
                         const unsigned* __restrict__ gcnt,
                         const unsigned* __restrict__ gmn,
                         const unsigned* __restrict__ gmx,
                         float* __restrict__ out) {
  int i = blockIdx.x * blockDim.x + threadIdx.x;
  if (i >= GROUPS) return;
  float s = 0.0f;
  unsigned c = 0u, mn = 0xFFFFFFFFu, mx = 0u;
#pragma unroll
  for (int r = 0; r < NCOPIES; ++r) {
    int o = r * GROUPS + i;
    s += gsum[o];
    c += gcnt[o];
    mn = min(mn, gmn[o]);
    mx = max(mx, gmx[o]);
  }
  out[i] = (float)(GROUPS - 1 - i);    // descending unique keys
  out[GROUPS + i] = s;                 // sum
  out[2 * GROUPS + i] = s / (float)c;  // mean
  out[3 * GROUPS + i] = unordf(mn);    // min
  out[4 * GROUPS + i] = unordf(mx);    // max
}

extern "C" void kernel_launch(void* const* d_in, const int* in_sizes, int n_in,
                              void* d_out, int out_size, void* d_ws,
                              size_t ws_size, hipStream_t stream) {
  const int* keys = (const int*)d_in[0];
  const float* vals = (const float*)d_in[1];
  int n = in_sizes[0];

  // d_ws layout: sum[NCOPIES*G] f32 | cnt | min | max  (128 KB total)
  float* gsum = (float*)d_ws;
  unsigned* gcnt = (unsigned*)d_ws + NCOPIES * GROUPS;
  unsigned* gmn = gcnt + NCOPIES * GROUPS;
  unsigned* gmx = gmn + NCOPIES * GROUPS;

  gb_init<<<(NCOPIES * GROUPS + BLOCK - 1) / BLOCK, BLOCK, 0, stream>>>(
      gsum, gcnt, gmn, gmx);
  gb_main<<<MAIN_GRID, BLOCK, 0, stream>>>(keys, vals, gsum, gcnt, gmn, gmx, n);
  gb_final<<<(GROUPS + BLOCK - 1) / BLOCK, BLOCK, 0, stream>>>(
      gsum, gcnt, gmn, gmx, (float*)d_out);
}